// AFG_20409684590889
// MI455X (gfx1250) — compile-verified
//
#include <hip/hip_runtime.h>

typedef __attribute__((ext_vector_type(2))) float v2f;
typedef __attribute__((ext_vector_type(8))) float v8f;

#define Bq 4
#define Cc 24
#define Hh 256
#define Ww 256
#define HWp (Hh * Ww)
#define C3 72
#define CK 216

// ---------------------------------------------------------------------------
// K1: fusion = 1x1 conv (24 <- 72) over concat(x, pha, amp), via WMMA f32.
// One wave handles 16 contiguous pixels (N-tile). M = 24 -> two 16-row tiles.
// K = 72 -> 18 WMMA K-steps of 4 (3 tensors x 6 steps). All global accesses
// use one base address per tensor + compile-time immediate offsets (<8MB).
// ---------------------------------------------------------------------------
__global__ __launch_bounds__(256) void k_fusion(
    const float* __restrict__ x, const float* __restrict__ pha,
    const float* __restrict__ amp, const float* __restrict__ wf,
    const float* __restrict__ bf, float* __restrict__ fusion) {
  __shared__ float s_wf[32 * C3];  // rows 24..31 zero-padded

  const int tid = threadIdx.x;
  for (int idx = tid; idx < 32 * C3; idx += 256) {
    int m = idx / C3, k = idx - m * C3;
    s_wf[idx] = (m < Cc) ? wf[m * C3 + k] : 0.f;
  }
  __syncthreads();

  const int wave = tid >> 5;
  const int lane = tid & 31;
  const int ln = lane & 15, hi = lane >> 4;
  const int p0 = (blockIdx.x * 8 + wave) * 16;  // 16 pixels per wave
  const int b  = p0 >> 16;                      // HWp == 65536
  const int sp = p0 & (HWp - 1);

  // LDS base offsets for A fragments (per-iteration offsets are immediates)
  const int abase0 = ln * C3 + hi * 2;
  const int abase1 = (16 + ln) * C3 + hi * 2;

  v8f acc0 = {0, 0, 0, 0, 0, 0, 0, 0};
  v8f acc1 = {0, 0, 0, 0, 0, 0, 0, 0};

  const float* bases[3] = {x, pha, amp};
#pragma unroll
  for (int t = 0; t < 3; ++t) {
    // one 64-bit base per tensor; channel offsets folded into immediates
    const float* qt = bases[t] + ((size_t)b * Cc + hi * 2) * HWp + sp + ln;
#pragma unroll
    for (int k6 = 0; k6 < 6; ++k6) {
      // B frag (4x16): vgpr0 = K=2*hi, vgpr1 = K=2*hi+1, col = ln
      v2f Bf;
      Bf.x = qt[k6 * 4 * HWp];
      Bf.y = qt[k6 * 4 * HWp + HWp];

      const int ch4 = (t * 6 + k6) * 4;  // compile-time column base
      v2f Af0, Af1;
      Af0.x = s_wf[abase0 + ch4];
      Af0.y = s_wf[abase0 + ch4 + 1];
      Af1.x = s_wf[abase1 + ch4];
      Af1.y = s_wf[abase1 + ch4 + 1];

      acc0 = __builtin_amdgcn_wmma_f32_16x16x4_f32(false, Af0, false, Bf,
                                                   (short)0, acc0, false,
                                                   false);
      acc1 = __builtin_amdgcn_wmma_f32_16x16x4_f32(false, Af1, false, Bf,
                                                   (short)0, acc1, false,
                                                   false);
    }
  }

  // D layout: row = tile*16 + v + 8*hi, col = ln. Stores from one base +
  // immediate offsets (v*HWp*4 <= 1.8 MB).
  float* st0 = fusion + ((size_t)b * Cc + hi * 8) * HWp + sp + ln;
#pragma unroll
  for (int v = 0; v < 8; ++v) st0[v * HWp] = acc0[v] + bf[v + hi * 8];
  if (hi == 0) {  // tile-1 rows 16..23 valid only for low half-wave
    float* st1 = fusion + ((size_t)b * Cc + 16) * HWp + sp + ln;
#pragma unroll
    for (int v = 0; v < 8; ++v) st1[v * HWp] = acc1[v] + bf[16 + v];
  }
}

// ---------------------------------------------------------------------------
// K2: spatial mean of fusion per (b,c). 96 blocks x 256 threads.
// ---------------------------------------------------------------------------
__global__ __launch_bounds__(256) void k_mean(const float* __restrict__ fusion,
                                              float* __restrict__ mf) {
  __shared__ float red[256];
  const int bc = blockIdx.x;
  const float* p = fusion + (size_t)bc * HWp;
  float s = 0.f;
  for (int i = threadIdx.x; i < HWp; i += 256) s += p[i];
  red[threadIdx.x] = s;
  __syncthreads();
  for (int o = 128; o > 0; o >>= 1) {
    if (threadIdx.x < (unsigned)o) red[threadIdx.x] += red[threadIdx.x + o];
    __syncthreads();
  }
  if (threadIdx.x == 0) mf[bc] = red[0] * (1.0f / (float)HWp);
}

// ---------------------------------------------------------------------------
// K3: channel attention = conv1d(kernel 3, pad 1) over channel axis. Tiny.
// ---------------------------------------------------------------------------
__global__ void k_ca(const float* __restrict__ mf, const float* __restrict__ wca,
                     const float* __restrict__ bca, float* __restrict__ cav) {
  int t = threadIdx.x;
  if (t < Bq * Cc) {
    int b = t / Cc, c = t - b * Cc;
    float a = bca[0];
#pragma unroll
    for (int k = 0; k < 3; ++k) {
      int cc2 = c + k - 1;
      if (cc2 >= 0 && cc2 < Cc) a += wca[k] * mf[b * Cc + cc2];
    }
    cav[t] = a;
  }
}

// ---------------------------------------------------------------------------
// K4: fused att -> per-pixel kernel GEMM (WMMA) -> dynamic 3x3 filter apply.
// One block = 16x16 pixel tile, 256 threads (8 waves).
// ---------------------------------------------------------------------------
__global__ __launch_bounds__(256) void k_main(
    const float* __restrict__ x, const float* __restrict__ fusion,
    const float* __restrict__ cav, const float* __restrict__ wsa,
    const float* __restrict__ bsa, const float* __restrict__ wk,
    const float* __restrict__ bk, float* __restrict__ out) {
  __shared__ float s_fus[Cc * 324];  // fusion tile + 1-pixel halo (18x18)
  __shared__ float s_att[Cc * 256];  // att, laid out [ch][pixel]
  float* s_out = s_fus;              // reused after stage B (6400 <= 7776)

  const int b  = blockIdx.z;
  const int ox = blockIdx.x * 16 - 1;
  const int oy = blockIdx.y * 16 - 1;
  const int tid = threadIdx.x;

  // per-batch base pointers: 32-bit element offsets (max 6.3 MB) from here
  const float* xb = x + (size_t)b * Cc * HWp;
  const float* fb = fusion + (size_t)b * Cc * HWp;
  float* ob = out + (size_t)b * Cc * HWp;

  // Stage A: load fusion tile + halo (zero-padded at image border).
  // Also prefetch the x tile rows needed by stage C (global_prefetch_b8).
  for (int idx = tid; idx < Cc * 18; idx += 256) {
    int c = idx / 18, row = idx - c * 18;
    int hy = oy + row;
    hy = hy < 0 ? 0 : (hy > Hh - 1 ? Hh - 1 : hy);
    int hx = ox < 0 ? 0 : ox;
    __builtin_prefetch(&xb[(unsigned)(c * HWp + hy * Ww + hx)], 0, 3);
  }
  for (int idx = tid; idx < Cc * 324; idx += 256) {
    int c = idx / 324, h = idx - c * 324;
    int hy = oy + h / 18, hx = ox + (h - (h / 18) * 18);
    float v = 0.f;
    if (hy >= 0 && hy < Hh && hx >= 0 && hx < Ww)
      v = fb[(unsigned)(c * HWp + hy * Ww + hx)];
    s_fus[idx] = v;
  }
  __syncthreads();

  // Stage B: att = depthwise 3x3(fusion) + b_sa + ca  (one thread per pixel)
  {
    int py = tid >> 4, px = tid & 15;
    int hb = (py + 1) * 18 + (px + 1);
    for (int c = 0; c < Cc; ++c) {
      float a = bsa[c] + cav[b * Cc + c];
#pragma unroll
      for (int ki = 0; ki < 3; ++ki)
#pragma unroll
        for (int kj = 0; kj < 3; ++kj)
          a += wsa[c * 9 + ki * 3 + kj] *
               s_fus[c * 324 + hb + (ki - 1) * 18 + (kj - 1)];
      s_att[c * 256 + tid] = a;
    }
  }
  __syncthreads();

  // zero output accumulator (aliases s_fus; dead rows add 0 to clamped row c=23)
  for (int c = 0; c < Cc + 1; ++c) s_out[c * 256 + tid] = 0.f;
  __syncthreads();

  // Stage C: kers = W_kernel[216,24] x att (WMMA) ; combine with unfold(x)
  const int wave = tid >> 5, lane = tid & 31;
  const int ln = lane & 15, hi = lane >> 4;

  // Preload all B fragments (att) for this wave's 2 pixel groups: registers.
  v2f Bf[2][6];
#pragma unroll
  for (int g = 0; g < 2; ++g) {
    int p = (wave * 2 + g) * 16 + ln;
#pragma unroll
    for (int kkq = 0; kkq < 6; ++kkq) {
      int ch = kkq * 4 + hi * 2;
      Bf[g][kkq].x = s_att[ch * 256 + p];
      Bf[g][kkq].y = s_att[(ch + 1) * 256 + p];
    }
  }

  for (int mt = 0; mt < 14; ++mt) {  // 216 rows -> 14 M-tiles of 16
    // A fragments shared by both pixel groups; branchless zero-pad row >=216
    const int m = mt * 16 + ln;
    const int mc = m < CK ? m : 0;
    v2f Af[6];
#pragma unroll
    for (int kkq = 0; kkq < 6; ++kkq) {
      int ch = kkq * 4 + hi * 2;
      float a0 = wk[mc * Cc + ch];
      float a1 = wk[mc * Cc + ch + 1];
      Af[kkq].x = (m < CK) ? a0 : 0.f;
      Af[kkq].y = (m < CK) ? a1 : 0.f;
    }

    v8f acc0 = {0, 0, 0, 0, 0, 0, 0, 0};
    v8f acc1 = {0, 0, 0, 0, 0, 0, 0, 0};
#pragma unroll
    for (int kkq = 0; kkq < 6; ++kkq) {
      acc0 = __builtin_amdgcn_wmma_f32_16x16x4_f32(false, Af[kkq], false,
                                                   Bf[0][kkq], (short)0, acc0,
                                                   false, false);
      acc1 = __builtin_amdgcn_wmma_f32_16x16x4_f32(false, Af[kkq], false,
                                                   Bf[1][kkq], (short)0, acc1,
                                                   false, false);
    }

    // per-row metadata: identical for both pixel groups -> hoist
    int rvalid[8], cidx[8], kio[8], kjo[8];
    float bkv[8];
#pragma unroll
    for (int v = 0; v < 8; ++v) {
      int r  = mt * 16 + v + hi * 8;  // kernel-channel row
      int rc = r < CK ? r : CK - 1;   // clamped (dead rows add 0)
      rvalid[v] = (r < CK);
      bkv[v] = bk[rc];
      int c = rc / 9;
      int kidx = rc - c * 9;
      cidx[v] = c;
      kio[v] = kidx / 3;
      kjo[v] = kidx - kio[v] * 3;
    }

#pragma unroll
    for (int g = 0; g < 2; ++g) {
      const v8f acc = (g == 0) ? acc0 : acc1;
      const int g2 = wave * 2 + g;
      const int p  = g2 * 16 + ln;
      const int gy = blockIdx.y * 16 + g2;
      const int gx = blockIdx.x * 16 + ln;
#pragma unroll
      for (int v = 0; v < 8; ++v) {
        float ker = acc[v] + bkv[v];
        int yy = gy + kio[v] - 1, xx = gx + kjo[v] - 1;
        int yyc = yy < 0 ? 0 : (yy > Hh - 1 ? Hh - 1 : yy);
        int xxc = xx < 0 ? 0 : (xx > Ww - 1 ? Ww - 1 : xx);
        float xv = xb[(unsigned)(cidx[v] * HWp + yyc * Ww + xxc)];
        bool ok = rvalid[v] & (yy >= 0) & (yy < Hh) & (xx >= 0) & (xx < Ww);
        float add = ok ? ker * xv : 0.f;
        atomicAdd(&s_out[cidx[v] * 256 + p], add);  // ds_add_f32, branchless
      }
    }
  }
  __syncthreads();

  // write output tile
  {
    int py = tid >> 4, px = tid & 15;
    int gy = blockIdx.y * 16 + py, gx = blockIdx.x * 16 + px;
    for (int c = 0; c < Cc; ++c)
      ob[(unsigned)(c * HWp + gy * Ww + gx)] = s_out[c * 256 + tid];
  }
}

// ---------------------------------------------------------------------------
extern "C" void kernel_launch(void* const* d_in, const int* in_sizes, int n_in,
                              void* d_out, int out_size, void* d_ws,
                              size_t ws_size, hipStream_t stream) {
  const float* x   = (const float*)d_in[0];
  const float* pha = (const float*)d_in[1];
  const float* amp = (const float*)d_in[2];
  const float* wf  = (const float*)d_in[3];
  const float* bf  = (const float*)d_in[4];
  const float* wsa = (const float*)d_in[5];
  const float* bsa = (const float*)d_in[6];
  const float* wca = (const float*)d_in[7];
  const float* bca = (const float*)d_in[8];
  const float* wk  = (const float*)d_in[9];
  const float* bk  = (const float*)d_in[10];
  float* out = (float*)d_out;

  float* fusion = (float*)d_ws;                    // B*C*H*W floats (25.2 MB)
  float* mf     = fusion + (size_t)Bq * Cc * HWp;  // 96 floats
  float* cav    = mf + Bq * Cc;                    // 96 floats

  // K1: 262144 pixels / (8 waves * 16 px) = 2048 blocks
  k_fusion<<<2048, 256, 0, stream>>>(x, pha, amp, wf, bf, fusion);
  k_mean<<<Bq * Cc, 256, 0, stream>>>(fusion, mf);
  k_ca<<<1, 128, 0, stream>>>(mf, wca, bca, cav);
  dim3 grid(Ww / 16, Hh / 16, Bq);
  k_main<<<grid, 256, 0, stream>>>(x, fusion, cav, wsa, bsa, wk, bk, out);
}